// TensorProductLayer_12549894439658
// MI455X (gfx1250) — compile-verified
//
#include <hip/hip_runtime.h>
#include <math.h>

// CDNA5 wave32 WMMA operand types: f32 16x16x4 -> A,B = 2 VGPRs/lane, C/D = 8 VGPRs/lane
typedef __attribute__((ext_vector_type(2))) float v2f;
typedef __attribute__((ext_vector_type(8))) float v8f;

#define C_CH 32           // channels (hardcoded in reference)
#define NB 5              // num RBF basis
#define WAVES_PER_BLK 8   // 256 threads / wave32

// RBF constants: SPREAD = 3.5/5 = 0.7 ; centers = 0, 0.875, 1.75, 2.625, 3.5
#define SPREAD 0.7f
#define CSTEP  0.875f
#define INV_SQRT3 0.57735026918962576f
#define INV_SQRT2 0.70710678118654752f

struct TileRegs {
    v8f rad[4][2];   // [coeff set][channel tile], D-matrix fragments
};

__global__ __launch_bounds__(256) void tp_edge_kernel(
    const float* __restrict__ nodes0,    // [N, C]
    const float* __restrict__ nodes1,    // [N, C, 3]
    const float* __restrict__ edges,     // [E, 3]
    const int*   __restrict__ senders,   // [E]
    const int*   __restrict__ receivers, // [E]
    const float* __restrict__ c00, const float* __restrict__ c01,
    const float* __restrict__ c10, const float* __restrict__ c11,  // [C, NB]
    const float* __restrict__ w000, const float* __restrict__ w011,
    const float* __restrict__ w101, const float* __restrict__ w110,
    const float* __restrict__ w111,                                 // [C]
    float* __restrict__ out0,            // [N, 2C]
    float* __restrict__ out1,            // [N, 3C, 3]
    int E)
{
    const int lane = threadIdx.x & 31;
    const int wv   = threadIdx.x >> 5;
    const int tile = blockIdx.x * WAVES_PER_BLK + wv;  // 16 edges per wave
    const int e0   = tile * 16;
    const int half = lane >> 4;    // 0: lanes 0-15 (K 0,1 / rows M 0-7), 1: lanes 16-31 (K 2,3 / rows M 8-15)
    const int lm   = lane & 15;    // A row (edge) index / B,D column (channel) index
    const float hsel = half ? 0.0f : 1.0f;   // forces unconditional loads of K=4 operands

    // Stage this wave's 16 edges (vec + sender + receiver) in LDS so every lane
    // can iterate its 8 D-matrix rows in the message phase.
    __shared__ float sX[WAVES_PER_BLK][16], sY[WAVES_PER_BLK][16], sZ[WAVES_PER_BLK][16];
    __shared__ int   sS[WAVES_PER_BLK][16], sR[WAVES_PER_BLK][16];

    // ---- RBF phase: both halves compute rbf for edge (e0 + lane%16) ----
    int eIdx = e0 + lm;
    int eCl  = (eIdx < E) ? eIdx : (E - 1);          // clamp (OOB rows masked later)
    float ex = edges[(size_t)eCl * 3 + 0];
    float ey = edges[(size_t)eCl * 3 + 1];
    float ez = edges[(size_t)eCl * 3 + 2];
    if (half == 0) {
        sX[wv][lm] = ex; sY[wv][lm] = ey; sZ[wv][lm] = ez;
        sS[wv][lm] = senders[eCl];
        sR[wv][lm] = receivers[eCl];
    }

    float r = sqrtf(ex * ex + ey * ey + ez * ez);
    float rb[NB];
    #pragma unroll
    for (int b = 0; b < NB; ++b) {
        float d = r - CSTEP * (float)b;
        rb[b] = __expf(-SPREAD * d * d);
    }

    // A operands (16x4 f32, ISA layout): vgpr0 = K=2*half, vgpr1 = K=2*half+1.
    // K=5 padded to 8 -> second WMMA carries K=4 in slot 0 (half 0) and zeros elsewhere.
    v2f a0, a1;
    a0.x = half ? rb[2] : rb[0];
    a0.y = half ? rb[3] : rb[1];
    a1.x = rb[4] * hsel;
    a1.y = 0.0f;

    // ---- WMMA phase: rad[s][t] = rbf(16x8-padded) x coeffs_s^T (8x16), tile t of channels ----
    const float* csets[4] = {c00, c01, c10, c11};
    TileRegs tr;
    #pragma unroll
    for (int s = 0; s < 4; ++s) {
        #pragma unroll
        for (int t = 0; t < 2; ++t) {
            const float* row = csets[s] + (size_t)(t * 16 + lm) * NB; // B[k][n] = coeffs[n][k]
            // Unconditional loads of the whole 5-value row (branch-free B build).
            const float k0 = row[0], k1 = row[1], k2 = row[2], k3 = row[3], k4 = row[4];
            v2f b0, b1;
            b0.x = half ? k2 : k0;        // K = 0 or 2
            b0.y = half ? k3 : k1;        // K = 1 or 3
            b1.x = k4 * hsel;             // K = 4 (K=6 -> pad 0)
            b1.y = 0.0f;                  // K = 5 / 7 -> pad 0
            v8f acc = {0.f, 0.f, 0.f, 0.f, 0.f, 0.f, 0.f, 0.f};
            acc = __builtin_amdgcn_wmma_f32_16x16x4_f32(
                false, a0, false, b0, (short)0, acc, false, false);
            acc = __builtin_amdgcn_wmma_f32_16x16x4_f32(
                false, a1, false, b1, (short)0, acc, false, false);
            tr.rad[s][t] = acc;
        }
    }

    __syncthreads();

    // ---- Message + scatter phase ----
    // D layout: lane L, reg g -> edge row m = g + 8*(L/16), channel n = t*16 + (L%16)
    auto message_phase = [&](bool check) __attribute__((always_inline)) {
        #pragma unroll
        for (int t = 0; t < 2; ++t) {
            const int c = t * 16 + lm;
            const float W000 = w000[c], W011 = w011[c], W101 = w101[c];
            const float W110 = w110[c], W111 = w111[c];
            #pragma unroll
            for (int g = 0; g < 8; ++g) {
                const int m = g + 8 * half;
                if (check && (e0 + m >= E)) continue;
                const float x = sX[wv][m], y = sY[wv][m], z = sZ[wv][m];
                const int snd = sS[wv][m], rcv = sR[wv][m];

                const float n0v = nodes0[(size_t)snd * C_CH + c];
                const float* n1p = nodes1 + ((size_t)snd * C_CH + c) * 3;
                const float n1x = n1p[0], n1y = n1p[1], n1z = n1p[2];

                const float r00 = tr.rad[0][t][g];
                const float r01 = tr.rad[1][t][g];
                const float r10 = tr.rad[2][t][g];
                const float r11 = tr.rad[3][t][g];

                // l x l_f -> l_o depthwise tensor products
                const float m000 = W000 * n0v * r00;                          // 0x0->0
                const float dotv = n1x * x + n1y * y + n1z * z;
                const float m110 = W110 * r11 * dotv * INV_SQRT3;             // 1x1->0
                const float a01  = W011 * n0v * r01;                          // 0x1->1 (scale of evec)
                const float a10  = W101 * r10;                                // 1x0->1 (scale of n1)
                const float k2   = W111 * r11 * INV_SQRT2;                    // 1x1->1 (scale of cross)
                const float cx = n1y * z - n1z * y;
                const float cy = n1z * x - n1x * z;
                const float cz = n1x * y - n1y * x;

                float* o0 = out0 + (size_t)rcv * (2 * C_CH);
                atomicAdd(o0 + c, m000);
                atomicAdd(o0 + C_CH + c, m110);

                float* o1 = out1 + (size_t)rcv * (3 * C_CH) * 3;
                float* p011 = o1 + (size_t)c * 3;
                atomicAdd(p011 + 0, a01 * x);
                atomicAdd(p011 + 1, a01 * y);
                atomicAdd(p011 + 2, a01 * z);
                float* p101 = o1 + ((size_t)C_CH + c) * 3;
                atomicAdd(p101 + 0, a10 * n1x);
                atomicAdd(p101 + 1, a10 * n1y);
                atomicAdd(p101 + 2, a10 * n1z);
                float* p111 = o1 + ((size_t)2 * C_CH + c) * 3;
                atomicAdd(p111 + 0, k2 * cx);
                atomicAdd(p111 + 1, k2 * cy);
                atomicAdd(p111 + 2, k2 * cz);
            }
        }
    };

    if (e0 + 16 <= E) {
        message_phase(false);   // full tile: branch-free hot path
    } else {
        message_phase(true);    // partial tail tile only
    }
}

extern "C" void kernel_launch(void* const* d_in, const int* in_sizes, int n_in,
                              void* d_out, int out_size, void* d_ws, size_t ws_size,
                              hipStream_t stream) {
    const float* nodes0    = (const float*)d_in[0];
    const float* nodes1    = (const float*)d_in[1];
    const float* edges     = (const float*)d_in[2];
    const int*   senders   = (const int*)d_in[3];
    const int*   receivers = (const int*)d_in[4];
    const float* c00  = (const float*)d_in[5];
    const float* c01  = (const float*)d_in[6];
    const float* c10  = (const float*)d_in[7];
    const float* c11  = (const float*)d_in[8];
    const float* w000 = (const float*)d_in[9];
    const float* w011 = (const float*)d_in[10];
    const float* w101 = (const float*)d_in[11];
    const float* w110 = (const float*)d_in[12];
    const float* w111 = (const float*)d_in[13];

    const int N = in_sizes[0] / C_CH;  // nodes0 = N*C
    const int E = in_sizes[3];         // senders = E

    float* out0 = (float*)d_out;                         // [N, 2C]
    float* out1 = out0 + (size_t)N * 2 * C_CH;           // [N, 3C, 3]

    // Zero the scatter targets (graph-capture-safe memset node).
    hipMemsetAsync(d_out, 0, (size_t)out_size * sizeof(float), stream);

    const int tiles  = (E + 15) / 16;                    // one wave32 per 16 edges
    const int blocks = (tiles + WAVES_PER_BLK - 1) / WAVES_PER_BLK;
    tp_edge_kernel<<<blocks, WAVES_PER_BLK * 32, 0, stream>>>(
        nodes0, nodes1, edges, senders, receivers,
        c00, c01, c10, c11,
        w000, w011, w101, w110, w111,
        out0, out1, E);
}